// GraphConvBlock_11871289606705
// MI455X (gfx1250) — compile-verified
//
#include <hip/hip_runtime.h>
#include <math.h>

// ---------------------------------------------------------------------------
// GraphConv + top-2 MoE FFN + BatchNorm for MI455X (gfx1250, wave32, WMMA).
// All GEMMs run on v_wmma_f32_16x16x4_f32 (full fp32, matches fp32 reference;
// bf16 WMMA would risk flipping the top-2 gate argmax).
// Each wave computes TWO 16-row tiles per B tile -> one pair of B loads feeds
// two WMMAs (halves weight-load instructions and L2 weight traffic).
// FFN: 32 rows/block, h1 streamed through LDS in two 256-col halves (64 KB);
// input rows staged with GLOBAL_LOAD_ASYNC_TO_LDS_B128 + s_wait_asynccnt.
// ---------------------------------------------------------------------------

#define N_NODES   16384
#define N_EDGES   262144
#define D_IN      256
#define D_MODEL   256
#define H_INNER   512
#define N_EXPERTS 64
#define CAP       1024
#define N_PAIRS   (N_NODES * 2)
#define BN_EPS    1e-5f

typedef __attribute__((ext_vector_type(2))) float v2f;
typedef __attribute__((ext_vector_type(8))) float v8f;
typedef int v4i_vs __attribute__((vector_size(16)));   // matches builtin param type

// D = A(16x4) * B(4x16) + C ; fp32 WMMA
#define WMMA4(a, b, c) \
  __builtin_amdgcn_wmma_f32_16x16x4_f32(false, (a), false, (b), (short)0, (c), false, false)

#if defined(__HIP_DEVICE_COMPILE__) && __has_builtin(__builtin_amdgcn_global_load_async_to_lds_b128)
#define HAVE_ASYNC_LDS 1
#else
#define HAVE_ASYNC_LDS 0
#endif

__device__ __forceinline__ void wait_async_zero() {
#if defined(__HIP_DEVICE_COMPILE__)
#if __has_builtin(__builtin_amdgcn_s_wait_asynccnt)
  __builtin_amdgcn_s_wait_asynccnt(0);
#else
  asm volatile("s_wait_asynccnt 0" ::: "memory");
#endif
#endif
}

// ---------------------------------------------------------------------------
// Utility kernels
// ---------------------------------------------------------------------------
__global__ void fill_zero_kernel(float* __restrict__ p, int n) {
  int i = blockIdx.x * blockDim.x + threadIdx.x;
  int stride = gridDim.x * blockDim.x;
  for (; i < n; i += stride) p[i] = 0.0f;
}

__global__ void degree_kernel(const int* __restrict__ src, const int* __restrict__ dst,
                              float* __restrict__ deg_out, float* __restrict__ deg_in) {
  int i = blockIdx.x * blockDim.x + threadIdx.x;
  if (i < N_EDGES) {
    atomicAdd(&deg_out[src[i]], 1.0f);   // integer-valued fp adds: exact, order-free
    atomicAdd(&deg_in[dst[i]], 1.0f);
  }
}

__global__ void rsqrt_inplace_kernel(float* __restrict__ p, int n) {
  int i = blockIdx.x * blockDim.x + threadIdx.x;
  if (i < n) p[i] = rsqrtf(fmaxf(p[i], 1.0f));
}

// h = feats * rsqrt(deg_out)[row]
__global__ void scale_rows_kernel(const float* __restrict__ feats,
                                  const float* __restrict__ rsd,
                                  float* __restrict__ h) {
  const int row = blockIdx.x, c = threadIdx.x;
  h[row * D_IN + c] = feats[row * D_IN + c] * rsd[row];
}

// m[dst] += h[src]   (one edge per block; contiguous 1KB rows, coalesced)
__global__ void edge_scatter_kernel(const int* __restrict__ src, const int* __restrict__ dst,
                                    const float* __restrict__ h, float* __restrict__ m) {
  const int e = blockIdx.x;
  const int s = src[e], d = dst[e];
  atomicAdd(&m[d * D_MODEL + threadIdx.x], h[s * D_MODEL + threadIdx.x]);
}

// ---------------------------------------------------------------------------
// x = (m * rsd_in) @ W_gc + feats @ W_res + (b_gc + b_res)
// one wave = 32 rows x 16 cols (two 16x16 tiles sharing every B tile)
// ---------------------------------------------------------------------------
__global__ void __launch_bounds__(256) x_gemm_kernel(
    const float* __restrict__ m, const float* __restrict__ rsd_in,
    const float* __restrict__ feats,
    const float* __restrict__ Wgc, const float* __restrict__ bgc,
    const float* __restrict__ Wres, const float* __restrict__ bres,
    float* __restrict__ x) {
  const int wid   = (blockIdx.x * 256 + threadIdx.x) >> 5;
  const int mpair = wid >> 4;          // 512 row-pair tiles (32 rows each)
  const int ntile = wid & 15;          // 16 col tiles
  const int row0 = mpair * 32, col0 = ntile * 16;
  const int lane = threadIdx.x & 31;
  const int half = lane >> 4, l16 = lane & 15;
  const float rs0 = rsd_in[row0 + l16];
  const float rs1 = rsd_in[row0 + 16 + l16];
  const int arow0 = (row0 + l16) * D_MODEL;
  const int arow1 = arow0 + 16 * D_MODEL;
  v8f acc0 = {}, acc1 = {};
  for (int k = 0; k < D_MODEL; k += 4) {
    v2f b;
    b.x = Wgc[(k + 2 * half) * D_MODEL + col0 + l16];
    b.y = Wgc[(k + 2 * half + 1) * D_MODEL + col0 + l16];
    v2f a0 = *reinterpret_cast<const v2f*>(&m[arow0 + k + 2 * half]);
    v2f a1 = *reinterpret_cast<const v2f*>(&m[arow1 + k + 2 * half]);
    a0.x *= rs0; a0.y *= rs0;
    a1.x *= rs1; a1.y *= rs1;
    acc0 = WMMA4(a0, b, acc0);
    acc1 = WMMA4(a1, b, acc1);
  }
  for (int k = 0; k < D_MODEL; k += 4) {
    v2f b;
    b.x = Wres[(k + 2 * half) * D_MODEL + col0 + l16];
    b.y = Wres[(k + 2 * half + 1) * D_MODEL + col0 + l16];
    v2f a0 = *reinterpret_cast<const v2f*>(&feats[arow0 + k + 2 * half]);
    v2f a1 = *reinterpret_cast<const v2f*>(&feats[arow1 + k + 2 * half]);
    acc0 = WMMA4(a0, b, acc0);
    acc1 = WMMA4(a1, b, acc1);
  }
  const int col = col0 + l16;
  const float bias = bgc[col] + bres[col];
  #pragma unroll
  for (int r = 0; r < 8; ++r) {
    x[(row0 + r + 8 * half) * D_MODEL + col]      = acc0[r] + bias;
    x[(row0 + 16 + r + 8 * half) * D_MODEL + col] = acc1[r] + bias;
  }
}

// ---------------------------------------------------------------------------
// logits = x @ Wg + bg   (N x 64, K = 256); 32 rows x 16 cols per wave
// ---------------------------------------------------------------------------
__global__ void __launch_bounds__(256) gate_gemm_kernel(
    const float* __restrict__ x, const float* __restrict__ Wg,
    const float* __restrict__ bg, float* __restrict__ logits) {
  const int wid   = (blockIdx.x * 256 + threadIdx.x) >> 5;
  const int mpair = wid >> 2;          // 512 row-pair tiles
  const int ntile = wid & 3;           // 4 col tiles
  const int row0 = mpair * 32, col0 = ntile * 16;
  const int lane = threadIdx.x & 31;
  const int half = lane >> 4, l16 = lane & 15;
  const int arow0 = (row0 + l16) * D_MODEL;
  const int arow1 = arow0 + 16 * D_MODEL;
  v8f acc0 = {}, acc1 = {};
  for (int k = 0; k < D_MODEL; k += 4) {
    v2f b;
    b.x = Wg[(k + 2 * half) * N_EXPERTS + col0 + l16];
    b.y = Wg[(k + 2 * half + 1) * N_EXPERTS + col0 + l16];
    v2f a0 = *reinterpret_cast<const v2f*>(&x[arow0 + k + 2 * half]);
    v2f a1 = *reinterpret_cast<const v2f*>(&x[arow1 + k + 2 * half]);
    acc0 = WMMA4(a0, b, acc0);
    acc1 = WMMA4(a1, b, acc1);
  }
  const int col = col0 + l16;
  const float bias = bg[col];
  #pragma unroll
  for (int r = 0; r < 8; ++r) {
    logits[(row0 + r + 8 * half) * N_EXPERTS + col]      = acc0[r] + bias;
    logits[(row0 + 16 + r + 8 * half) * N_EXPERTS + col] = acc1[r] + bias;
  }
}

// ---------------------------------------------------------------------------
// top-2 + softmax over selected logits; expert histogram
// ---------------------------------------------------------------------------
__global__ void topk_kernel(const float* __restrict__ logits, int* __restrict__ pair_e,
                            float* __restrict__ pair_g, int* __restrict__ counts) {
  const int t = blockIdx.x * blockDim.x + threadIdx.x;
  if (t >= N_NODES) return;
  const float* L = logits + t * N_EXPERTS;
  float v0 = -INFINITY, v1 = -INFINITY;
  int i0 = 0, i1 = 0;
  for (int e = 0; e < N_EXPERTS; ++e) {
    float v = L[e];
    if (v > v0)      { v1 = v0; i1 = i0; v0 = v; i0 = e; }
    else if (v > v1) { v1 = v;  i1 = e; }
  }
  const float e1  = expf(v1 - v0);          // stable: v1 <= v0
  const float inv = 1.0f / (1.0f + e1);
  pair_e[2 * t]     = i0;  pair_e[2 * t + 1] = i1;
  pair_g[2 * t]     = inv; pair_g[2 * t + 1] = e1 * inv;
  atomicAdd(&counts[i0], 1);
  atomicAdd(&counts[i1], 1);
}

__global__ void prefix_kernel(const int* __restrict__ counts, int* __restrict__ offs) {
  if (threadIdx.x == 0) {
    int run = 0;
    for (int e = 0; e < N_EXPERTS; ++e) { offs[e] = run; run += min(counts[e], CAP); }
  }
}

// assign slot (compacted per-expert rows) and copy x row into dispatch buffer
__global__ void dispatch_kernel(const int* __restrict__ pair_e, const float* __restrict__ x,
                                const int* __restrict__ offs, int* __restrict__ cursor,
                                int* __restrict__ pair_slot, float* __restrict__ buf) {
  __shared__ int s_slot;
  const int p = blockIdx.x;
  if (threadIdx.x == 0) {
    const int e = pair_e[p];
    const int r = atomicAdd(&cursor[e], 1);
    const int slot = (r < CAP) ? (offs[e] + r) : -1;
    pair_slot[p] = slot;
    s_slot = slot;
  }
  __syncthreads();
  const int slot = s_slot;
  if (slot >= 0)
    buf[slot * D_MODEL + threadIdx.x] = x[(p >> 1) * D_MODEL + threadIdx.x];
}

// ---------------------------------------------------------------------------
// Fused expert FFN: out = gelu(A @ w1[e] + b1[e]) @ w2[e] + b2[e]
// 32 rows per block; h1 streamed through LDS in two 256-column halves so
// layer-2 accumulates each K-chunk immediately (64 KB LDS total).
// Each wave: 2 n-tiles x 2 m-tiles -> every B-tile load feeds two WMMAs.
// Input rows staged via async global->LDS DMA (no VGPR round trip).
// ---------------------------------------------------------------------------
__global__ void __launch_bounds__(256) moe_ffn_kernel(
    const float* __restrict__ buf, const int* __restrict__ counts,
    const int* __restrict__ offs,
    const float* __restrict__ w1, const float* __restrict__ b1,
    const float* __restrict__ w2, const float* __restrict__ b2,
    float* __restrict__ outbuf) {
  __shared__ float sA[32 * D_MODEL];   // 32 KB : input rows
  __shared__ float sH[32 * 256];       // 32 KB : one 256-col half of h1
  const int e = blockIdx.x;
  const int cnt = min(counts[e], CAP);
  const int tile0 = blockIdx.y * 32;
  if (tile0 >= cnt) return;            // uniform exit: EXEC stays full for WMMA
  const int rowbase   = offs[e] + tile0;
  const int rows_here = cnt - tile0;
  const int rows_full = rows_here < 32 ? rows_here : 32;

#if HAVE_ASYNC_LDS
  // GLOBAL_LOAD_ASYNC_TO_LDS_B128: memory -> LDS without VGPR round trip.
  for (int i = threadIdx.x; i < 32 * (D_MODEL / 4); i += 256) {
    const int r  = i >> 6;             // 64 x b128 per row
    const int c4 = (i & 63) << 2;
    if (r < rows_full) {
      __builtin_amdgcn_global_load_async_to_lds_b128(
          (v4i_vs*)(buf + (size_t)(rowbase + r) * D_MODEL + c4),
          (v4i_vs*)(sA + r * D_MODEL + c4),
          0, 0);
    } else {
      *(float4*)(sA + r * D_MODEL + c4) = make_float4(0.f, 0.f, 0.f, 0.f);
    }
  }
  wait_async_zero();                   // ASYNCcnt not covered by __syncthreads
#else
  for (int i = threadIdx.x; i < 32 * (D_MODEL / 4); i += 256) {
    const int r  = i >> 6;
    const int c4 = (i & 63) << 2;
    float4 v = make_float4(0.f, 0.f, 0.f, 0.f);
    if (r < rows_full) v = *(const float4*)(buf + (size_t)(rowbase + r) * D_MODEL + c4);
    *(float4*)(sA + r * D_MODEL + c4) = v;
  }
#endif

  const int wave = threadIdx.x >> 5;
  const int lane = threadIdx.x & 31;
  const int half = lane >> 4, l16 = lane & 15;

  const float* w1e = w1 + (size_t)e * (D_MODEL * H_INNER);
  const float* b1e = b1 + e * H_INNER;
  const float* w2e = w2 + (size_t)e * (H_INNER * D_MODEL);
  const float* b2e = b2 + e * D_MODEL;

  v8f acc2[2][2] = {};                 // persistent layer-2 acc [n-tile j][m-tile]

  for (int hh = 0; hh < 2; ++hh) {
    __syncthreads();                   // sA ready / previous sH fully consumed
    // ---- layer 1: h1 columns [hh*256, hh*256+256) ----
    #pragma unroll
    for (int j = 0; j < 2; ++j) {
      const int nt   = wave * 2 + j;             // 0..15 inside this half
      const int colg = hh * 256 + nt * 16;       // global h1 column base
      v8f acc0 = {}, acc1 = {};
      for (int k = 0; k < D_MODEL; k += 4) {
        v2f b;
        b.x = w1e[(k + 2 * half) * H_INNER + colg + l16];
        b.y = w1e[(k + 2 * half + 1) * H_INNER + colg + l16];
        v2f a0 = *reinterpret_cast<const v2f*>(&sA[l16 * D_MODEL + k + 2 * half]);
        v2f a1 = *reinterpret_cast<const v2f*>(&sA[(16 + l16) * D_MODEL + k + 2 * half]);
        acc0 = WMMA4(a0, b, acc0);
        acc1 = WMMA4(a1, b, acc1);
      }
      const float bias = b1e[colg + l16];
      #pragma unroll
      for (int r = 0; r < 8; ++r) {
        float v0 = acc0[r] + bias;
        v0 = 0.5f * v0 * (1.0f + erff(v0 * 0.70710678118654752f));  // exact GELU
        sH[(r + 8 * half) * 256 + nt * 16 + l16] = v0;
        float v1 = acc1[r] + bias;
        v1 = 0.5f * v1 * (1.0f + erff(v1 * 0.70710678118654752f));
        sH[(16 + r + 8 * half) * 256 + nt * 16 + l16] = v1;
      }
    }
    __syncthreads();
    // ---- layer 2 partial: K-chunk [hh*256, hh*256+256) ----
    #pragma unroll
    for (int j = 0; j < 2; ++j) {
      const int col0 = (wave * 2 + j) * 16;
      for (int k = 0; k < 256; k += 4) {
        v2f b;
        b.x = w2e[(hh * 256 + k + 2 * half) * D_MODEL + col0 + l16];
        b.y = w2e[(hh * 256 + k + 2 * half + 1) * D_MODEL + col0 + l16];
        v2f a0 = *reinterpret_cast<const v2f*>(&sH[l16 * 256 + k + 2 * half]);
        v2f a1 = *reinterpret_cast<const v2f*>(&sH[(16 + l16) * 256 + k + 2 * half]);
        acc2[j][0] = WMMA4(a0, b, acc2[j][0]);
        acc2[j][1] = WMMA4(a1, b, acc2[j][1]);
      }
    }
  }

  #pragma unroll
  for (int j = 0; j < 2; ++j) {
    const int col = (wave * 2 + j) * 16 + l16;
    const float bias = b2e[col];
    #pragma unroll
    for (int mt = 0; mt < 2; ++mt) {
      #pragma unroll
      for (int r = 0; r < 8; ++r) {
        const int rl = mt * 16 + r + 8 * half;
        if (tile0 + rl < cnt)
          outbuf[(rowbase + rl) * D_MODEL + col] = acc2[j][mt][r] + bias;
      }
    }
  }
}

// y = x + g0*out[slot0] + g1*out[slot1]  (fixed order -> deterministic fp)
__global__ void combine_kernel(const float* __restrict__ x, const float* __restrict__ outbuf,
                               const int* __restrict__ pair_slot,
                               const float* __restrict__ pair_g, float* __restrict__ y) {
  const int t = blockIdx.x, c = threadIdx.x;
  float acc = x[t * D_MODEL + c];
  const int s0 = pair_slot[2 * t], s1 = pair_slot[2 * t + 1];
  if (s0 >= 0) acc += pair_g[2 * t]     * outbuf[s0 * D_MODEL + c];
  if (s1 >= 0) acc += pair_g[2 * t + 1] * outbuf[s1 * D_MODEL + c];
  y[t * D_MODEL + c] = acc;
}

// ---------------------------------------------------------------------------
// BatchNorm (training stats, biased var): deterministic two-level reduction
// ---------------------------------------------------------------------------
__global__ void bn_partial_kernel(const float* __restrict__ y,
                                  float* __restrict__ psum, float* __restrict__ psum2) {
  const int c = threadIdx.x;
  const int row0 = blockIdx.x * (N_NODES / 256);
  float s = 0.f, s2 = 0.f;
  for (int r = 0; r < N_NODES / 256; ++r) {
    const float v = y[(row0 + r) * D_MODEL + c];
    s += v; s2 += v * v;
  }
  psum [blockIdx.x * D_MODEL + c] = s;
  psum2[blockIdx.x * D_MODEL + c] = s2;
}

__global__ void bn_stats_kernel(const float* __restrict__ psum, const float* __restrict__ psum2,
                                float* __restrict__ meanArr, float* __restrict__ varArr) {
  const int c = threadIdx.x;
  float s = 0.f, s2 = 0.f;
  for (int b = 0; b < 256; ++b) { s += psum[b * D_MODEL + c]; s2 += psum2[b * D_MODEL + c]; }
  const float mean = s * (1.0f / N_NODES);
  meanArr[c] = mean;
  varArr[c]  = s2 * (1.0f / N_NODES) - mean * mean;
}

__global__ void bn_apply_kernel(const float* __restrict__ y, const float* __restrict__ meanArr,
                                const float* __restrict__ varArr, const float* __restrict__ gamma,
                                const float* __restrict__ beta, float* __restrict__ out) {
  const int t = blockIdx.x, c = threadIdx.x;
  out[t * D_MODEL + c] =
      (y[t * D_MODEL + c] - meanArr[c]) * rsqrtf(varArr[c] + BN_EPS) * gamma[c] + beta[c];
}

// ---------------------------------------------------------------------------
// Launcher
// ---------------------------------------------------------------------------
extern "C" void kernel_launch(void* const* d_in, const int* in_sizes, int n_in,
                              void* d_out, int out_size, void* d_ws, size_t ws_size,
                              hipStream_t stream) {
  (void)in_sizes; (void)n_in; (void)out_size; (void)ws_size;
  const float* feats = (const float*)d_in[0];
  const float* Wgc   = (const float*)d_in[1];
  const float* bgc   = (const float*)d_in[2];
  const float* Wres  = (const float*)d_in[3];
  const float* bres  = (const float*)d_in[4];
  const float* Wg    = (const float*)d_in[5];
  const float* bg    = (const float*)d_in[6];
  const float* w1    = (const float*)d_in[7];
  const float* b1    = (const float*)d_in[8];
  const float* w2    = (const float*)d_in[9];
  const float* b2    = (const float*)d_in[10];
  const float* gamma = (const float*)d_in[11];
  const float* beta  = (const float*)d_in[12];
  const int*   src   = (const int*)d_in[13];
  const int*   dst   = (const int*)d_in[14];
  float* out = (float*)d_out;

  char* ws = (char*)d_ws;
  float* deg_out   = (float*)(ws);                        // 64 KB (becomes rsd_out)
  float* deg_in    = (float*)(ws + (64u << 10));          // 64 KB (becomes rsd_in)
  int*   counts    = (int*)  (ws + (128u << 10));         // 64 i32
  int*   cursor    = counts + 64;
  int*   offs      = counts + 128;
  float* meanArr   = (float*)(ws + (132u << 10));
  float* varArr    = meanArr + 256;
  int*   pair_e    = (int*)  (ws + (136u << 10));         // 128 KB
  float* pair_g    = (float*)(ws + (264u << 10));         // 128 KB
  int*   pair_slot = (int*)  (ws + (392u << 10));         // 128 KB
  float* psum      = (float*)(ws + (520u << 10));         // 256 KB
  float* psum2     = (float*)(ws + (776u << 10));         // 256 KB
  float* logits    = (float*)(ws + (2ull << 20));         // 4 MB
  float* h         = (float*)(ws + (16ull << 20));        // 16 MB
  float* m         = (float*)(ws + (32ull << 20));        // 16 MB
  float* x         = (float*)(ws + (48ull << 20));        // 16 MB
  float* buf       = (float*)(ws + (64ull << 20));        // 32 MB (dispatch, compacted)
  float* outbuf    = (float*)(ws + (96ull << 20));        // 32 MB
  float* y         = h;                                   // h dead after edge scatter

  // zero-init accumulators
  fill_zero_kernel<<<64, 256, 0, stream>>>(deg_out, N_NODES);
  fill_zero_kernel<<<64, 256, 0, stream>>>(deg_in, N_NODES);
  fill_zero_kernel<<<1, 256, 0, stream>>>((float*)counts, 192);
  fill_zero_kernel<<<4096, 256, 0, stream>>>(m, N_NODES * D_MODEL);

  // GCN aggregation
  degree_kernel<<<N_EDGES / 256, 256, 0, stream>>>(src, dst, deg_out, deg_in);
  rsqrt_inplace_kernel<<<64, 256, 0, stream>>>(deg_out, N_NODES);
  rsqrt_inplace_kernel<<<64, 256, 0, stream>>>(deg_in, N_NODES);
  scale_rows_kernel<<<N_NODES, 256, 0, stream>>>(feats, deg_out, h);
  edge_scatter_kernel<<<N_EDGES, 256, 0, stream>>>(src, dst, h, m);

  // x = (m*rsd_in)@Wgc + feats@Wres + biases  (WMMA): 512 row-pairs x 16 cols
  x_gemm_kernel<<<1024, 256, 0, stream>>>(m, deg_in, feats, Wgc, bgc, Wres, bres, x);

  // gating (WMMA) + top-2
  gate_gemm_kernel<<<256, 256, 0, stream>>>(x, Wg, bg, logits);
  topk_kernel<<<N_NODES / 256, 256, 0, stream>>>(logits, pair_e, pair_g, counts);
  prefix_kernel<<<1, 64, 0, stream>>>(counts, offs);
  dispatch_kernel<<<N_PAIRS, 256, 0, stream>>>(pair_e, x, offs, cursor, pair_slot, buf);

  // expert FFN (WMMA, fused two layers through LDS, 32 rows per block)
  moe_ffn_kernel<<<dim3(N_EXPERTS, CAP / 32), 256, 0, stream>>>(
      buf, counts, offs, w1, b1, w2, b2, outbuf);

  // combine + BN
  combine_kernel<<<N_NODES, 256, 0, stream>>>(x, outbuf, pair_slot, pair_g, y);
  bn_partial_kernel<<<256, 256, 0, stream>>>(y, psum, psum2);
  bn_stats_kernel<<<1, 256, 0, stream>>>(psum, psum2, meanArr, varArr);
  bn_apply_kernel<<<N_NODES, 256, 0, stream>>>(y, meanArr, varArr, gamma, beta, out);
}